// MemristorDense_63685775065688
// MI455X (gfx1250) — compile-verified
//
#include <hip/hip_runtime.h>

// ---------------- problem constants ----------------
#define N_IN   1024
#define N_OUT  512
#define BATCH  128
#define M_ROWS (N_IN + 1)          // 1025 word lines (incl. bias row)
#define ISPLIT 8                   // i-dimension split for parallelism
#define SEG    129                 // ceil(1025/8)
#define CI     32                  // i-rows per LDS chunk in main kernel

// ---------------- CDNA5 feature probes ----------------
#if defined(__has_builtin)
# if __has_builtin(__builtin_amdgcn_global_load_async_to_lds_b128)
#  define HAVE_ASYNC_LDS 1
# endif
#endif

#ifdef HAVE_ASYNC_LDS
// builtin prototype (from hipcc diagnostic): (v4i AS1*, v4i AS3*, imm int, imm int)
typedef int v4i __attribute__((vector_size(16)));
typedef __attribute__((address_space(1))) v4i* g_v4i_ptr;
typedef __attribute__((address_space(3))) v4i* l_v4i_ptr;

__device__ __forceinline__ void wait_async0() {
# if __has_builtin(__builtin_amdgcn_s_wait_asynccnt)
  __builtin_amdgcn_s_wait_asynccnt(0);
# else
  asm volatile("s_wait_asynccnt 0x0" ::: "memory");
# endif
}
#endif

typedef __attribute__((ext_vector_type(2))) float v2f;
typedef __attribute__((ext_vector_type(8))) float v8f;

// ---------------- kernel 0: init the max slot ----------------
__global__ void kInit(unsigned* maxslot) { maxslot[0] = 0u; }

// ---------------- kernel 1: max |weights| over w_pos, w_neg, b_pos, b_neg ----
__global__ void kMax(const float* __restrict__ wp, const float* __restrict__ wn,
                     const float* __restrict__ bp, const float* __restrict__ bn,
                     unsigned* maxslot) {
  __shared__ float red[256];
  const int NT = 2 * 524288 + 2 * 512;
  float m = 0.0f;
  for (int idx = blockIdx.x * blockDim.x + threadIdx.x; idx < NT;
       idx += gridDim.x * blockDim.x) {
    float v;
    if (idx < 524288)       v = wp[idx];
    else if (idx < 1048576) v = wn[idx - 524288];
    else if (idx < 1049088) v = bp[idx - 1048576];
    else                    v = bn[idx - 1049088];
    m = fmaxf(m, fabsf(v));
  }
  red[threadIdx.x] = m;
  __syncthreads();
  for (int s = 128; s > 0; s >>= 1) {
    if (threadIdx.x < s) red[threadIdx.x] = fmaxf(red[threadIdx.x], red[threadIdx.x + s]);
    __syncthreads();
  }
  if (threadIdx.x == 0) atomicMax(maxslot, __float_as_uint(red[0]));  // floats >= 0: bit order == value order
}

// ---------------- kernel 2: fused coefficient table T[(i,j)] = (Ap, An, Ep, En) ---
// Exact fold: y-scale * V_REF * (k_G*w + G_MIN) = 0.5 * (w + 0.25*max_weight)
// E = log2(2.8 + 0.2 * noise)
__global__ void kTables(const float* __restrict__ wp_, const float* __restrict__ wn_,
                        const float* __restrict__ bp_, const float* __restrict__ bn_,
                        const float* __restrict__ noise, const float* __restrict__ maxslot,
                        float* __restrict__ T) {
  const int t = blockIdx.x * blockDim.x + threadIdx.x;
  if (t >= M_ROWS * N_OUT) return;
  const int i = t / N_OUT;
  const int j = t - i * N_OUT;
  const float wp = (i < N_IN) ? wp_[(size_t)i * N_OUT + j] : bp_[j];
  const float wn = (i < N_IN) ? wn_[(size_t)i * N_OUT + j] : bn_[j];
  const float c0 = 0.25f * maxslot[0];
  const float np = 2.8f + 0.2f * noise[(size_t)i * (2 * N_OUT) + 2 * j];
  const float nn = 2.8f + 0.2f * noise[(size_t)i * (2 * N_OUT) + 2 * j + 1];
  float4 o;
  o.x = 0.5f * (wp + c0);
  o.y = 0.5f * (wn + c0);
  o.z = __builtin_amdgcn_logf(np);   // v_log_f32 = log2
  o.w = __builtin_amdgcn_logf(nn);
  ((float4*)T)[t] = o;
}

// ---------------- kernel 3: per-(b,i) table SL = (sign(x), log2(2|x|)) --------
__global__ void kSL(const float* __restrict__ x, float* __restrict__ SL) {
  const int t = blockIdx.x * blockDim.x + threadIdx.x;
  if (t >= BATCH * M_ROWS) return;
  const int b = t / M_ROWS;
  const int i = t - b * M_ROWS;
  const float v = (i < N_IN) ? x[(size_t)b * N_IN + i] : 1.0f;  // bias line: ratio = 2
  float2 o;
  o.x = (v > 0.0f) ? 1.0f : ((v < 0.0f) ? -1.0f : 0.0f);
  o.y = 1.0f + __builtin_amdgcn_logf(fabsf(v));                 // log2(2|x|)
  ((float2*)SL)[t] = o;
}

// ---------------- kernel 4: main crossbar reduction --------------------------
// Block: 256 threads = 8 wave32. lane = output pair (32 pairs per block),
// wave = tile of 16 batches (8 waves cover all 128). grid = (16 pair tiles, 8 i-segments).
// Coefficient chunks staged global->LDS (async b128 when available), read once from HBM.
__global__ void kMain(const float* __restrict__ T, const float* __restrict__ SL,
                      float* __restrict__ part) {
  __shared__ __align__(16) float4 tile[CI * 32];      // 16 KB
  const int tid  = threadIdx.x;
  const int lane = tid & 31;
  const int wave = tid >> 5;
  const int j    = blockIdx.x * 32 + lane;            // output pair index
  const int z    = blockIdx.y;                        // i segment
  const int i0   = z * SEG;
  const int i1   = min(M_ROWS, i0 + SEG);
  const int b0   = wave * 16;

  float acc[16];
#pragma unroll
  for (int q = 0; q < 16; ++q) acc[q] = 0.0f;

  const float4* Tg  = (const float4*)T;
  const float2* SLg = (const float2*)SL;

  for (int ic = i0; ic < i1; ic += CI) {
    const int cn = min(CI, i1 - ic);
    // cooperative stage of the (cn x 32-pair) coefficient tile
    for (int e = tid; e < cn * 32; e += 256) {
      const int ci = e >> 5, jp = e & 31;
      const float4* g = Tg + ((size_t)(ic + ci) * N_OUT + blockIdx.x * 32 + jp);
#ifdef HAVE_ASYNC_LDS
      __builtin_amdgcn_global_load_async_to_lds_b128(
          (g_v4i_ptr)(const void*)g, (l_v4i_ptr)&tile[e], 0, 0);
#else
      tile[e] = *g;
#endif
    }
#ifdef HAVE_ASYNC_LDS
    wait_async0();
#endif
    __syncthreads();
    if (ic + CI < i1)  // pull next chunk toward L2 while we compute
      __builtin_prefetch(Tg + ((size_t)(ic + CI) * N_OUT + blockIdx.x * 32 + lane), 0, 3);

    for (int ci = 0; ci < cn; ++ci) {
      const float4 c = tile[ci * 32 + lane];          // Ap, An, Ep, En
      const float2* slp = SLg + ((size_t)b0 * M_ROWS + (ic + ci));  // lane-uniform
#pragma unroll
      for (int q = 0; q < 16; ++q) {
        const float2 sl = slp[(size_t)q * M_ROWS];    // (sign, log2 ratio), uniform per wave
        const float tP = __builtin_amdgcn_exp2f(c.z * sl.y);  // ratio^Ep
        const float tN = __builtin_amdgcn_exp2f(c.w * sl.y);  // ratio^En
        acc[q] = fmaf(sl.x, fmaf(c.x, tP, -(c.y * tN)), acc[q]);
      }
    }
    __syncthreads();
  }
#pragma unroll
  for (int q = 0; q < 16; ++q)
    part[((size_t)z * BATCH + (b0 + q)) * N_OUT + j] = acc[q];
}

// ---------------- kernel 5: reduce the 8 i-segment partials via WMMA ---------
// y[o] = sum_z part[z][o]. Ones(16x4) x B(4x16) with the 8 slices in the K dim
// across two chained v_wmma_f32_16x16x4_f32. Exactly 4096 waves, EXEC all-ones.
__global__ void kReduceWmma(const float* __restrict__ part, float* __restrict__ y) {
  const int tid    = blockIdx.x * blockDim.x + threadIdx.x;
  const int waveId = tid >> 5;
  const int lane   = tid & 31;
  const int o      = waveId * 16 + (lane & 15);
  const int zBase  = (lane < 16) ? 0 : 2;   // B layout: lanes 0-15 hold K=0,1; lanes 16-31 hold K=2,3
  v2f a; a.x = 1.0f; a.y = 1.0f;            // ones A-matrix: D[m,n] = sum_k B[k,n]
  v2f b0, b1;
  b0.x = part[(size_t)(zBase + 0) * (BATCH * N_OUT) + o];
  b0.y = part[(size_t)(zBase + 1) * (BATCH * N_OUT) + o];
  b1.x = part[(size_t)(zBase + 4) * (BATCH * N_OUT) + o];
  b1.y = part[(size_t)(zBase + 5) * (BATCH * N_OUT) + o];
  v8f c = {};
  c = __builtin_amdgcn_wmma_f32_16x16x4_f32(false, a, false, b0, (short)0, c, false, false);
  c = __builtin_amdgcn_wmma_f32_16x16x4_f32(false, a, false, b1, (short)0, c, false, false);
  if (lane < 16) y[o] = c[0];               // row M=0 lives in VGPR0, lanes 0-15
}

// ---------------- launcher ----------------
extern "C" void kernel_launch(void* const* d_in, const int* in_sizes, int n_in,
                              void* d_out, int out_size, void* d_ws, size_t ws_size,
                              hipStream_t stream) {
  (void)in_sizes; (void)n_in; (void)out_size; (void)ws_size;
  const float* x  = (const float*)d_in[0];
  const float* wp = (const float*)d_in[1];
  const float* wn = (const float*)d_in[2];
  const float* bp = (const float*)d_in[3];
  const float* bn = (const float*)d_in[4];
  const float* nz = (const float*)d_in[5];

  char* ws = (char*)d_ws;
  unsigned* maxslot = (unsigned*)ws;                                   // 4 B (256 B reserved)
  const size_t T_off    = 256;
  const size_t T_bytes  = (size_t)M_ROWS * N_OUT * 16;                 // 8,396,800 B
  const size_t SL_off   = T_off + T_bytes;
  const size_t SL_bytes = (size_t)BATCH * M_ROWS * 8;                  // 1,049,600 B
  const size_t P_off    = SL_off + SL_bytes;                           // 2 MB partials
  float* T    = (float*)(ws + T_off);
  float* SL   = (float*)(ws + SL_off);
  float* part = (float*)(ws + P_off);
  float* y    = (float*)d_out;

  kInit<<<1, 1, 0, stream>>>(maxslot);
  kMax<<<512, 256, 0, stream>>>(wp, wn, bp, bn, maxslot);
  kTables<<<(M_ROWS * N_OUT + 255) / 256, 256, 0, stream>>>(wp, wn, bp, bn, nz,
                                                            (const float*)maxslot, T);
  kSL<<<(BATCH * M_ROWS + 255) / 256, 256, 0, stream>>>(x, SL);
  kMain<<<dim3(N_OUT / 32, ISPLIT), 256, 0, stream>>>(T, SL, part);
  kReduceWmma<<<(BATCH * N_OUT * 2) / 256, 256, 0, stream>>>(part, y);
}